// GNNWithXGB_66262755442783
// MI455X (gfx1250) — compile-verified
//
#include <hip/hip_runtime.h>
#include <hip/hip_bf16.h>

// GCN (2-layer) for MI455X / gfx1250, wave32.
// Dense GEMMs use V_WMMA_F32_16X16X4_F32 (fp32-exact) with the self-loop +
// bias epilogue fused in (saves a full read+write pass over h per layer).
// Edge aggregation: coalesced gathers + global_atomic_add_f32 scatters; the
// whole working set (~80MB) is resident in the 192MB L2.

typedef __attribute__((ext_vector_type(2))) float v2f;
typedef __attribute__((ext_vector_type(8))) float v8f;

#define N_IN_DIM 64
#define N_HID_DIM 48
#define N_OUT_DIM 32

// ---------------- degree / norm ----------------

__global__ void __launch_bounds__(256)
k_deg_init(float* __restrict__ deg, int n) {
    int i = blockIdx.x * 256 + threadIdx.x;
    if (i < n) deg[i] = 1.0f;           // self-loop contributes 1 to every node
}

__global__ void __launch_bounds__(256)
k_deg_count(const long long* __restrict__ ei, float* __restrict__ deg, int n_edges) {
    int e = blockIdx.x * 256 + threadIdx.x;
    if (e < n_edges) {
        int dst = (int)ei[(size_t)n_edges + e];
        atomicAdd(&deg[dst], 1.0f);
    }
}

__global__ void __launch_bounds__(256)
k_rsqrt_inplace(float* __restrict__ d, int n) {
    int i = blockIdx.x * 256 + threadIdx.x;
    if (i < n) d[i] = rsqrtf(d[i]);     // deg >= 1 always (self-loops)
}

// ------- fused GEMM + self-loop/bias epilogue -------
// H   = (RELU?)A[n,K] @ W[K,NOUT]
// AGG = bias + dinv^2 * H        (aggregation buffer initialized in-register)
//
// One wave -> one 16x16 tile. K stepped by 4 with V_WMMA_F32_16X16X4_F32.
// A-frag (16x4 f32): lane m=lane&15 is row M=m; lanes<16 hold K=k,k+1,
// lanes>=16 hold K=k+2,k+3 (one float2 load per lane).
// B-frag (4x16): row-striped across lanes, served from LDS-resident W.
// D/C (16x16 f32, 8 VGPRs): vgpr v: lanes<16 -> M=v, lanes>=16 -> M=v+8; N=lane&15.

template <int K, int NOUT, bool RELU>
__global__ void __launch_bounds__(128)
k_gemm_fused(const float* __restrict__ A, const float* __restrict__ W,
             const float* __restrict__ bias, const float* __restrict__ dinv,
             float* __restrict__ H, float* __restrict__ AGG, int n_rows) {
    __shared__ float sW[K * NOUT];      // <= 12KB of the 320KB WGP LDS

    // cooperative stage of W into LDS (before any wave-uniform early exit)
    for (int i = threadIdx.x; i < K * NOUT; i += 128) sW[i] = W[i];
    __syncthreads();

    const int lane = threadIdx.x & 31;
    const int wave = threadIdx.x >> 5;
    constexpr int TPR = NOUT >> 4;                      // tiles per row
    const int n_tiles = (n_rows >> 4) * TPR;            // n_rows % 16 == 0
    const int tile = blockIdx.x * 4 + wave;
    if (tile >= n_tiles) return;                        // uniform per wave

    const int rt = tile / TPR;
    const int ct = tile - rt * TPR;
    const int rbase = rt << 4;
    const int cbase = ct << 4;

    const int m    = lane & 15;
    const int koff = (lane >> 4) << 1;                  // 0 or 2

    v8f c = {};
    const float* arow = A + (size_t)(rbase + m) * K + koff;

#pragma unroll
    for (int k = 0; k < K; k += 4) {
        float2 av = *reinterpret_cast<const float2*>(arow + k);
        v2f a, b;
        a.x = RELU ? (av.x < 0.f ? 0.f : av.x) : av.x;
        a.y = RELU ? (av.y < 0.f ? 0.f : av.y) : av.y;
        b.x = sW[(k + koff)     * NOUT + cbase + m];
        b.y = sW[(k + koff + 1) * NOUT + cbase + m];
        // (neg_a, A, neg_b, B, c_mod, C, reuse_a, reuse_b)
        c = __builtin_amdgcn_wmma_f32_16x16x4_f32(false, a, false, b,
                                                  (short)0, c, false, false);
    }

    // fused epilogue: write h and the self-loop+bias-initialized agg buffer
    const int rowadd = (lane >> 4) << 3;                // 0 or 8
    const float bv = bias[cbase + m];
    const size_t obase = (size_t)(rbase + rowadd) * NOUT + cbase + m;
#pragma unroll
    for (int v = 0; v < 8; ++v) {
        float hv = c[v];
        float d  = dinv[rbase + rowadd + v];            // lane-uniform per half-wave
        H  [obase + (size_t)v * NOUT] = hv;
        AGG[obase + (size_t)v * NOUT] = bv + d * d * hv;
    }
}

// ---------------- edge aggregation: out[dst] += dinv[src]*dinv[dst]*h[src] ----
// 16 lanes per edge; lane l covers features l, l+16, (l+32). Coalesced gathers,
// global_atomic_add_f32 scatters (L2-resident working set).

__global__ void __launch_bounds__(256)
k_edge_agg(const long long* __restrict__ ei, const float* __restrict__ h,
           const float* __restrict__ dinv, float* __restrict__ out,
           int n_edges, int F) {
    int gid = blockIdx.x * 256 + threadIdx.x;
    int e = gid >> 4;
    int l = gid & 15;
    if (e >= n_edges) return;
    int src = (int)ei[e];
    int dst = (int)ei[(size_t)n_edges + e];
    float w = dinv[src] * dinv[dst];
    const float* hs = h + (size_t)src * F;
    float* od = out + (size_t)dst * F;
    for (int f = l; f < F; f += 16)
        atomicAdd(&od[f], w * hs[f]);
}

// ---------------- launch ----------------

static inline size_t align256(size_t x) { return (x + 255) & ~(size_t)255; }

extern "C" void kernel_launch(void* const* d_in, const int* in_sizes, int n_in,
                              void* d_out, int out_size, void* d_ws, size_t ws_size,
                              hipStream_t stream) {
    const float*     x  = (const float*)d_in[0];
    const long long* ei = (const long long*)d_in[1];   // int64 edge_index [2, E]
    const float*     W1 = (const float*)d_in[2];
    const float*     b1 = (const float*)d_in[3];
    const float*     W2 = (const float*)d_in[4];
    const float*     b2 = (const float*)d_in[5];
    float*           out = (float*)d_out;

    const int n  = in_sizes[0] / N_IN_DIM;   // 100000 (multiple of 16)
    const int ne = in_sizes[1] / 2;          // 1600000

    // workspace: [dinv: n] [h: n*48] [agg: n*48]; layer-2 h (n*32) reuses h slot
    char* ws = (char*)d_ws;
    float* dinv = (float*)ws;
    float* h    = (float*)(ws + align256((size_t)n * 4));
    float* agg  = (float*)(ws + align256((size_t)n * 4) + align256((size_t)n * N_HID_DIM * 4));

    // degrees -> dinv (in place)
    k_deg_init<<<(n + 255) / 256, 256, 0, stream>>>(dinv, n);
    k_deg_count<<<(ne + 255) / 256, 256, 0, stream>>>(ei, dinv, ne);
    k_rsqrt_inplace<<<(n + 255) / 256, 256, 0, stream>>>(dinv, n);

    // layer 1: h = x@W1 ; agg = b1 + dinv^2*h (fused) ; scatter edges into agg
    {
        int tiles = (n >> 4) * (N_HID_DIM >> 4);
        k_gemm_fused<N_IN_DIM, N_HID_DIM, false>
            <<<(tiles + 3) / 4, 128, 0, stream>>>(x, W1, b1, dinv, h, agg, n);
        k_edge_agg<<<((size_t)ne * 16 + 255) / 256, 256, 0, stream>>>(ei, h, dinv, agg, ne, N_HID_DIM);
    }

    // layer 2: h2 = relu(agg)@W2 (h2 reuses h slot) ; out = b2 + dinv^2*h2 (fused) ; scatter
    {
        int tiles = (n >> 4) * (N_OUT_DIM >> 4);
        k_gemm_fused<N_HID_DIM, N_OUT_DIM, true>
            <<<(tiles + 3) / 4, 128, 0, stream>>>(agg, W2, b2, dinv, h, out, n);
        k_edge_agg<<<((size_t)ne * 16 + 255) / 256, 256, 0, stream>>>(ei, h, dinv, out, ne, N_OUT_DIM);
    }
}